// Attention_2_43739946942864
// MI455X (gfx1250) — compile-verified
//
#include <hip/hip_runtime.h>
#include <cmath>

typedef float v2f __attribute__((ext_vector_type(2)));
typedef float v8f __attribute__((ext_vector_type(8)));

namespace {
constexpr int kB = 4, kH = 8, kS = 2048, kD = 64;
constexpr int kRowPad = 2052;          // 2048 + 4 floats: de-conflicts 64-bank LDS
constexpr float kNegBig = -1e9f;
}

__global__ __launch_bounds__(256)
void attn_fused(const float* __restrict__ Q, const float* __restrict__ K,
                const float* __restrict__ V, const int* __restrict__ M,
                float* __restrict__ Out, float* __restrict__ P)
{
  __shared__ float sc[16 * kRowPad];     // 131,328 B score strip
  __shared__ float oacc_lds[16 * kD];    // 4 KB output accumulator
  __shared__ float red[256];
  __shared__ float rowmax[16];
  __shared__ float rowinv[16];

  const int tid  = threadIdx.x;
  const int lane = tid & 31;
  const int wave = tid >> 5;
  const int n    = lane & 15;            // N (cols of B/C) or M (rows of A)
  const int half = lane >> 4;
  const int koff = half * 2;             // K offset inside a 4-wide K chunk

  const int qtile = blockIdx.x & (kS / 16 - 1);   // 128 strips per (b,h)
  const int bh    = blockIdx.x >> 7;
  const int b     = bh / kH;
  const int q0    = qtile * 16;

  const float* Qb = Q + ((size_t)bh * kS + q0) * kD;
  const float* Kb = K + (size_t)bh * kS * kD;
  const float* Vb = V + (size_t)bh * kS * kD;
  const int*   Mb = M + ((size_t)b * kS + q0) * kS;

  for (int i = tid; i < 16 * kD; i += 256) oacc_lds[i] = 0.0f;

  // Preload the 16x64 Q strip as 16 f32 A-fragments (contiguous float2/lane).
  v2f qf[16];
  #pragma unroll
  for (int kk = 0; kk < 16; ++kk)
    qf[kk] = *(const v2f*)(Qb + n * kD + kk * 4 + koff);

  // ---- Phase 1: scores = exp(Q K^T) / sqrt(64), masked fill, stash in LDS ----
  for (int j = wave; j < kS / 16; j += 8) {
    v8f acc = {};
    const float* kb = Kb + ((size_t)(j * 16 + n)) * kD + koff;
    #pragma unroll
    for (int kk = 0; kk < 16; ++kk) {
      v2f bf = *(const v2f*)(kb + kk * 4);
      acc = __builtin_amdgcn_wmma_f32_16x16x4_f32(false, qf[kk], false, bf,
                                                  (short)0, acc, false, false);
    }
    #pragma unroll
    for (int r = 0; r < 8; ++r) {
      const int row = r + half * 8;                 // C/D layout: VGPR r -> M=r (+8)
      float s = expf(acc[r]) * 0.125f;              // exp(logit)/sqrt(d)
      if (Mb[(size_t)row * kS + j * 16 + n] == 0) s = kNegBig;
      sc[row * kRowPad + j * 16 + n] = s;
    }
  }
  __syncthreads();

  // ---- Phase 2: per-row max and softmax denominator ----
  const int srow = tid >> 4, scol = tid & 15;
  float m = -3.0e38f;
  for (int c = scol; c < kS; c += 16) m = fmaxf(m, sc[srow * kRowPad + c]);
  red[tid] = m;
  __syncthreads();
  if (tid < 16) {
    float mm = red[tid * 16];
    #pragma unroll
    for (int i = 1; i < 16; ++i) mm = fmaxf(mm, red[tid * 16 + i]);
    rowmax[tid] = mm;
  }
  __syncthreads();
  const float rm = rowmax[srow];
  float ssum = 0.0f;
  for (int c = scol; c < kS; c += 16) ssum += expf(sc[srow * kRowPad + c] - rm);
  red[tid] = ssum;
  __syncthreads();
  if (tid < 16) {
    float t = 0.0f;
    #pragma unroll
    for (int i = 0; i < 16; ++i) t += red[tid * 16 + i];
    rowinv[tid] = 1.0f / t;
  }
  __syncthreads();

  // ---- Phase 3a: normalize in LDS, stream p_attn to HBM (non-temporal) ----
  const float inv = rowinv[srow];
  float* prow = P + ((size_t)bh * kS + q0 + srow) * kS;
  for (int c = scol; c < kS; c += 16) {
    float pv = expf(sc[srow * kRowPad + c] - rm) * inv;
    sc[srow * kRowPad + c] = pv;
    __builtin_nontemporal_store(pv, prow + c);
  }
  __syncthreads();

  // ---- Phase 3b: Out = P @ V  (P from LDS, V from L2) ----
  v8f oacc[4] = {v8f{}, v8f{}, v8f{}, v8f{}};
  for (int j = wave; j < kS / 16; j += 8) {
    v2f af[4];
    #pragma unroll
    for (int kk = 0; kk < 4; ++kk)       // A fragment: P[m][j*16 + 4kk + koff..+1]
      af[kk] = *(const v2f*)&sc[n * kRowPad + j * 16 + kk * 4 + koff];
    #pragma unroll
    for (int nt = 0; nt < 4; ++nt) {     // D = 64 -> 4 N-tiles of 16
      #pragma unroll
      for (int kk = 0; kk < 4; ++kk) {
        const int kg = j * 16 + kk * 4 + koff;
        const float* vp = Vb + (size_t)kg * kD + nt * 16 + n;
        v2f bf;
        bf.x = vp[0];
        bf.y = vp[kD];
        oacc[nt] = __builtin_amdgcn_wmma_f32_16x16x4_f32(false, af[kk], false, bf,
                                                         (short)0, oacc[nt],
                                                         false, false);
      }
    }
  }
  #pragma unroll
  for (int nt = 0; nt < 4; ++nt) {
    #pragma unroll
    for (int r = 0; r < 8; ++r)
      atomicAdd(&oacc_lds[(r + half * 8) * kD + nt * 16 + n], oacc[nt][r]);
  }
  __syncthreads();

  {
    const int d0 = scol * 4;
    float* op = Out + ((size_t)bh * kS + q0 + srow) * kD + d0;
    #pragma unroll
    for (int i = 0; i < 4; ++i) op[i] = oacc_lds[srow * kD + d0 + i];
  }
}

extern "C" void kernel_launch(void* const* d_in, const int* in_sizes, int n_in,
                              void* d_out, int out_size, void* d_ws, size_t ws_size,
                              hipStream_t stream) {
  const float* q = (const float*)d_in[0];
  const float* k = (const float*)d_in[1];
  const float* v = (const float*)d_in[2];
  const int*   m = (const int*)d_in[3];
  float* out = (float*)d_out;
  float* p   = out + (size_t)kB * kH * kS * kD;   // tuple order: (out, p_attn)
  dim3 grid(kB * kH * (kS / 16));                 // 4096 workgroups
  attn_fused<<<grid, 256, 0, stream>>>(q, k, v, m, out, p);
}